// SMPLKinematics_39127152066802
// MI455X (gfx1250) — compile-verified
//
#include <hip/hip_runtime.h>

#define B_BATCH 1024
#define NUM_TF  24
#define NV      6890
#define N3      20670   // V*3
#define NPAD    20672   // N3 padded to 16
#define KTOT    220     // 207 PF + 1 pad + 10 betas + 2 pad
#define KPAIRS  110     // KTOT/2
#define BETA    10
#define NTILE_N2 646    // ceil(20672/32): two 16-wide N tiles per wave
#define NTILE_V 431     // ceil(6890/16)

typedef __attribute__((ext_vector_type(2))) float v2f;
typedef __attribute__((ext_vector_type(8))) float v8f;

__constant__ int c_parents[NUM_TF] = {-1,0,0,0,1,2,3,4,5,6,7,8,9,9,9,12,13,14,16,17,18,19,20,21};

// ---------------------------------------------------------------------------
// Kernel 0: build pair-packed fused B' operand for the blendshape GEMM.
//   B'[r][n]: r<207 -> posedirs[r,n]; r==207 -> 0; 208..217 -> shapedirs[n,r-208];
//             218..219 -> 0; n >= N3 -> 0.
//   Packed:  Bp[p*NPAD + n] = (B'[2p][n], B'[2p+1][n])  as float2 (v2f)
// ---------------------------------------------------------------------------
__device__ __forceinline__ float bprime(const float* __restrict__ posedirs,
                                        const float* __restrict__ shapedirs,
                                        int r, int n) {
  if (n >= N3) return 0.f;
  if (r < 207) return posedirs[r * N3 + n];
  if (r >= 208 && r < 218) return shapedirs[n * BETA + (r - 208)];
  return 0.f;
}

__global__ void pack_b_kernel(const float* __restrict__ posedirs,
                              const float* __restrict__ shapedirs,
                              float* __restrict__ Bp) {
  int p = blockIdx.y;                                   // 0..109 row pair
  int n = blockIdx.x * blockDim.x + threadIdx.x;        // 0..NPAD-1
  if (n >= NPAD) return;
  float b0 = bprime(posedirs, shapedirs, 2 * p, n);
  float b1 = bprime(posedirs, shapedirs, 2 * p + 1, n);
  v2f* out = (v2f*)Bp;
  v2f val; val.x = b0; val.y = b1;
  out[p * NPAD + n] = val;
}

// ---------------------------------------------------------------------------
// Kernel 1: joint regressor buffers
// ---------------------------------------------------------------------------
__global__ void joints_kernel(const float* __restrict__ Jreg,
                              const float* __restrict__ vtemp,
                              const float* __restrict__ shapedirs,
                              float* __restrict__ jsd,
                              float* __restrict__ jt) {
  int idx = blockIdx.x * blockDim.x + threadIdx.x;
  if (idx >= NUM_TF * 33) return;
  int j = idx / 33, s = idx % 33;
  const float* row = Jreg + j * NV;
  float acc = 0.f;
  if (s < 30) {
    int i = s / 10, d = s % 10;
    for (int v = 0; v < NV; ++v) acc += row[v] * shapedirs[(v * 3 + i) * BETA + d];
    jsd[j * 30 + s] = acc;
  } else {
    int i = s - 30;
    for (int v = 0; v < NV; ++v) acc += row[v] * vtemp[v * 3 + i];
    jt[j * 3 + i] = acc;
  }
}

// ---------------------------------------------------------------------------
// Kernel 2: per (b, joint): Rodrigues rotation, unposed joint, packed PF rows.
//   PFp (pair-packed [KPAIRS][B] of float2): element for (row r, batch b) is
//   float index (r>>1)*B*2 + b*2 + (r&1).
// ---------------------------------------------------------------------------
__global__ void rot_kernel(const float* __restrict__ pose,
                           const float* __restrict__ betas,
                           const float* __restrict__ jsd,
                           const float* __restrict__ jt,
                           float* __restrict__ dR,
                           float* __restrict__ dJ,
                           float* __restrict__ PFp) {
  int idx = blockIdx.x * blockDim.x + threadIdx.x;
  if (idx >= B_BATCH * NUM_TF) return;
  int b = idx / NUM_TF, k = idx % NUM_TF;

  float tx = pose[b * 72 + k * 3 + 0];
  float ty = pose[b * 72 + k * 3 + 1];
  float tz = pose[b * 72 + k * 3 + 2];
  float ang = sqrtf(tx * tx + ty * ty + tz * tz + 1e-12f);
  float inv = 1.0f / ang;
  float ax = tx * inv, ay = ty * inv, az = tz * inv;
  float s = sinf(ang), c = cosf(ang);
  float t = 1.0f - c;

  float R[9];
  R[0] = c + t * ax * ax;      R[1] = t * ax * ay - s * az; R[2] = t * ax * az + s * ay;
  R[3] = t * ax * ay + s * az; R[4] = c + t * ay * ay;      R[5] = t * ay * az - s * ax;
  R[6] = t * ax * az - s * ay; R[7] = t * ay * az + s * ax; R[8] = c + t * az * az;

  float* rout = dR + idx * 9;
#pragma unroll
  for (int e = 0; e < 9; ++e) rout[e] = R[e];

  if (k >= 1) {
    int base = (k - 1) * 9;
#pragma unroll
    for (int e = 0; e < 9; ++e) {
      float iv = (e == 0 || e == 4 || e == 8) ? 1.f : 0.f;
      int r = base + e;
      PFp[(r >> 1) * (B_BATCH * 2) + b * 2 + (r & 1)] = R[e] - iv;
    }
  } else {
    PFp[(207 >> 1) * (B_BATCH * 2) + b * 2 + 1] = 0.f;   // row 207
#pragma unroll
    for (int d = 0; d < BETA; ++d) {
      int r = 208 + d;
      PFp[(r >> 1) * (B_BATCH * 2) + b * 2 + (r & 1)] = betas[b * BETA + d];
    }
    PFp[(218 >> 1) * (B_BATCH * 2) + b * 2 + 0] = 0.f;   // row 218
    PFp[(219 >> 1) * (B_BATCH * 2) + b * 2 + 1] = 0.f;   // row 219
  }

  // unposed joint location: jt + jsd @ betas
#pragma unroll
  for (int i = 0; i < 3; ++i) {
    float acc = jt[k * 3 + i];
    for (int d = 0; d < BETA; ++d) acc += jsd[k * 30 + i * 10 + d] * betas[b * BETA + d];
    dJ[(b * NUM_TF + k) * 3 + i] = acc;
  }
}

// ---------------------------------------------------------------------------
// Kernel 3: forward kinematics chain, one thread per batch, chain in LDS.
// Emits A[b,k] in K-pair-interleaved layout for the skin GEMM B operand:
//   Ap[((b*12 + k/2)*16 + c)*2 + (k&1)] = A[b,k,c]
// ---------------------------------------------------------------------------
__global__ __launch_bounds__(32) void fk_kernel(const float* __restrict__ dR,
                                                const float* __restrict__ dJ,
                                                float* __restrict__ Ap) {
  __shared__ float G[32][NUM_TF * 12];  // per-thread chain: 9 rot + 3 trans
  int tid = threadIdx.x;
  int b = blockIdx.x * 32 + tid;
  if (b >= B_BATCH) return;

  for (int k = 0; k < NUM_TF; ++k) {
    float R[9], trel[3];
    const float* rp = dR + (b * NUM_TF + k) * 9;
#pragma unroll
    for (int e = 0; e < 9; ++e) R[e] = rp[e];
    int p = c_parents[k];
#pragma unroll
    for (int i = 0; i < 3; ++i) {
      float jv = dJ[(b * NUM_TF + k) * 3 + i];
      trel[i] = (p < 0) ? jv : (jv - dJ[(b * NUM_TF + p) * 3 + i]);
    }
    float* gc = G[tid] + k * 12;
    if (p < 0) {
#pragma unroll
      for (int e = 0; e < 9; ++e) gc[e] = R[e];
#pragma unroll
      for (int i = 0; i < 3; ++i) gc[9 + i] = trel[i];
    } else {
      float* gp = G[tid] + p * 12;
#pragma unroll
      for (int i = 0; i < 3; ++i) {
#pragma unroll
        for (int j = 0; j < 3; ++j)
          gc[i * 3 + j] = gp[i * 3 + 0] * R[0 + j] + gp[i * 3 + 1] * R[3 + j] +
                          gp[i * 3 + 2] * R[6 + j];
        gc[9 + i] = gp[9 + i] + gp[i * 3 + 0] * trel[0] + gp[i * 3 + 1] * trel[1] +
                    gp[i * 3 + 2] * trel[2];
      }
    }
  }

  for (int k = 0; k < NUM_TF; ++k) {
    float* gc = G[tid] + k * 12;
    float jv[3];
#pragma unroll
    for (int i = 0; i < 3; ++i) jv[i] = dJ[(b * NUM_TF + k) * 3 + i];
    float row[16];
#pragma unroll
    for (int i = 0; i < 3; ++i) {
      float off = gc[i * 3 + 0] * jv[0] + gc[i * 3 + 1] * jv[1] + gc[i * 3 + 2] * jv[2];
      row[i * 4 + 0] = gc[i * 3 + 0];
      row[i * 4 + 1] = gc[i * 3 + 1];
      row[i * 4 + 2] = gc[i * 3 + 2];
      row[i * 4 + 3] = gc[9 + i] - off;
    }
    row[12] = 0.f; row[13] = 0.f; row[14] = 0.f; row[15] = 1.f;
    float* ab = Ap + ((b * 12 + (k >> 1)) * 16) * 2 + (k & 1);
#pragma unroll
    for (int c = 0; c < 16; ++c) ab[c * 2] = row[c];
  }
}

// ---------------------------------------------------------------------------
// Kernel 4: fused pose+shape blendshape GEMM via V_WMMA_F32_16X16X4_F32.
// Per wave: one 16(M=batch) x 32(N) tile = 2 accumulators sharing the A load.
// Inner loop: 3 x global_load_b64 + 2 x wmma, branch-free.
// ---------------------------------------------------------------------------
__global__ __launch_bounds__(256) void posed_gemm_kernel(
    const float* __restrict__ PFp,
    const float* __restrict__ Bp,
    const float* __restrict__ vtemp,
    float* __restrict__ out) {
  int lane = threadIdx.x & 31;
  int wave = threadIdx.x >> 5;
  int nt2 = blockIdx.x * 8 + wave;       // wave-uniform
  if (nt2 >= NTILE_N2) return;           // whole wave exits: EXEC stays full
  int m0 = blockIdx.y * 16;
  int h = lane >> 4;                     // half-wave
  int col = lane & 15;
  int n0 = nt2 * 32 + col;
  int n1 = n0 + 16;
  int mrow = m0 + col;

  float tv0 = (n0 < N3) ? vtemp[n0] : 0.f;
  float tv1 = (n1 < N3) ? vtemp[n1] : 0.f;
  v8f acc0, acc1;
#pragma unroll
  for (int r = 0; r < 8; ++r) { acc0[r] = tv0; acc1[r] = tv1; }

  const v2f* ap  = (const v2f*)PFp + h * B_BATCH + mrow;
  const v2f* bp0 = (const v2f*)Bp + h * NPAD + n0;
  const v2f* bp1 = bp0 + 16;

#pragma unroll 2
  for (int s = 0; s < KPAIRS / 2; ++s) {   // 55 K-steps of 4
    v2f a  = ap[0];
    v2f b0 = bp0[0];
    v2f b1 = bp1[0];
    ap  += 2 * B_BATCH;
    bp0 += 2 * NPAD;
    bp1 += 2 * NPAD;
    acc0 = __builtin_amdgcn_wmma_f32_16x16x4_f32(false, a, false, b0,
                                                 (short)0, acc0, false, false);
    acc1 = __builtin_amdgcn_wmma_f32_16x16x4_f32(false, a, false, b1,
                                                 (short)0, acc1, false, false);
  }

  int mbase = m0 + (h << 3);               // rows: vgpr r -> m = mbase + r
  if (n0 < N3) {
#pragma unroll
    for (int r = 0; r < 8; ++r) out[(mbase + r) * N3 + n0] = acc0[r];
  }
  if (n1 < N3) {
#pragma unroll
    for (int r = 0; r < 8; ++r) out[(mbase + r) * N3 + n1] = acc1[r];
  }
}

// ---------------------------------------------------------------------------
// Kernel 5: LBS. Per wave: 16-vertex tile of T = weights(16x24) @ A[b](24x16)
// via 6 WMMAs (both operands single b64 loads), spill T tile to LDS, then 16
// lanes skin one vertex each, in place in d_out.
// ---------------------------------------------------------------------------
__global__ __launch_bounds__(256) void skin_kernel(
    const float* __restrict__ weights,
    const float* __restrict__ Ap,
    const float* __restrict__ trans,
    float* __restrict__ out) {
  __shared__ float Ts[8][16][17];
  int lane = threadIdx.x & 31;
  int wave = threadIdx.x >> 5;
  int vtile = blockIdx.x * 8 + wave;     // wave-uniform
  int b = blockIdx.y;
  bool active = (vtile < NTILE_V);

  if (active) {
    int col = lane & 15;
    int h = lane >> 4;
    int vrow = vtile * 16 + col;
    int vc = (vrow < NV) ? vrow : (NV - 1);
    const v2f* Abv = (const v2f*)Ap + (b * 12) * 16 + col;  // + p*16 per pair

    v8f acc;
#pragma unroll
    for (int r = 0; r < 8; ++r) acc[r] = 0.f;

#pragma unroll
    for (int kk = 0; kk < 24; kk += 4) {
      int ka = kk + h * 2;                 // even
      v2f a  = *(const v2f*)&weights[vc * NUM_TF + ka];
      v2f bb = Abv[(ka >> 1) * 16];
      acc = __builtin_amdgcn_wmma_f32_16x16x4_f32(false, a, false, bb,
                                                  (short)0, acc, false, false);
    }

    int mbase = h << 3;
#pragma unroll
    for (int r = 0; r < 8; ++r) Ts[wave][mbase + r][col] = acc[r];
  }

  __syncthreads();   // reached by all 256 threads

  if (active && lane < 16) {
    int v = vtile * 16 + lane;
    if (v < NV) {
      float* vp = out + b * N3 + v * 3;
      float vx = vp[0], vy = vp[1], vz = vp[2];
      const float* t = &Ts[wave][lane][0];
      float x = t[0] * vx + t[1] * vy + t[2] * vz + t[3];
      float y = t[4] * vx + t[5] * vy + t[6] * vz + t[7];
      float z = t[8] * vx + t[9] * vy + t[10] * vz + t[11];
      float w = t[12] * vx + t[13] * vy + t[14] * vz + t[15];
      float invw = 1.0f / w;
      vp[0] = x * invw + trans[b * 3 + 0];
      vp[1] = y * invw + trans[b * 3 + 1];
      vp[2] = z * invw + trans[b * 3 + 2];
    }
  }
}

// ---------------------------------------------------------------------------
extern "C" void kernel_launch(void* const* d_in, const int* in_sizes, int n_in,
                              void* d_out, int out_size, void* d_ws, size_t ws_size,
                              hipStream_t stream) {
  const float* betas     = (const float*)d_in[0];
  const float* pose      = (const float*)d_in[1];
  const float* trans     = (const float*)d_in[2];
  const float* Jreg      = (const float*)d_in[3];
  const float* vtemp     = (const float*)d_in[4];
  const float* posedirs  = (const float*)d_in[5];
  const float* shapedirs = (const float*)d_in[6];
  const float* weights   = (const float*)d_in[7];
  float* out = (float*)d_out;

  // workspace layout (~22 MB of floats, all segment bases 8B-aligned)
  float* ws   = (float*)d_ws;
  float* dAp  = ws;                               // B*24*16          = 393216
  float* dPFp = dAp + B_BATCH * NUM_TF * 16;      // KPAIRS*B*2       = 225280
  float* dBp  = dPFp + KPAIRS * B_BATCH * 2;      // KPAIRS*NPAD*2    = 4547840
  float* dR   = dBp + KPAIRS * NPAD * 2;          // B*24*9           = 221184
  float* dJ   = dR + B_BATCH * NUM_TF * 9;        // B*24*3           = 73728
  float* djsd = dJ + B_BATCH * NUM_TF * 3;        // 24*30            = 720
  float* djt  = djsd + NUM_TF * 30;               // 24*3             = 72

  pack_b_kernel<<<dim3((NPAD + 255) / 256, KPAIRS), 256, 0, stream>>>(
      posedirs, shapedirs, dBp);
  joints_kernel<<<dim3((NUM_TF * 33 + 255) / 256), 256, 0, stream>>>(
      Jreg, vtemp, shapedirs, djsd, djt);
  rot_kernel<<<dim3((B_BATCH * NUM_TF + 255) / 256), 256, 0, stream>>>(
      pose, betas, djsd, djt, dR, dJ, dPFp);
  fk_kernel<<<dim3(B_BATCH / 32), 32, 0, stream>>>(dR, dJ, dAp);
  posed_gemm_kernel<<<dim3((NTILE_N2 + 7) / 8, B_BATCH / 16), 256, 0, stream>>>(
      dPFp, dBp, vtemp, out);
  skin_kernel<<<dim3((NTILE_V + 7) / 8, B_BATCH), 256, 0, stream>>>(
      weights, dAp, trans, out);
}